// GATNet_34033320853660
// MI455X (gfx1250) — compile-verified
//
#include <hip/hip_runtime.h>
#include <hip/hip_bf16.h>

#define NN 50000           // nodes
#define NE 800000          // raw edges
#define ET 850000          // edges + self loops
#define HID 64
#define OUTD 40
#define NEG_SLOPE 0.2f

typedef float v2f __attribute__((ext_vector_type(2)));
typedef float v8f __attribute__((ext_vector_type(8)));

// ---- monotone float <-> uint encoding for exact atomic max ----
__device__ __forceinline__ unsigned f2ord(float f) {
  unsigned u = __float_as_uint(f);
  return (u & 0x80000000u) ? ~u : (u | 0x80000000u);
}
__device__ __forceinline__ float ord2f(unsigned u) {
  return (u & 0x80000000u) ? __uint_as_float(u & 0x7fffffffu)
                           : __uint_as_float(~u);
}

__device__ __forceinline__ int esrc(const int* ei, int e) { return e < NE ? ei[e] : e - NE; }
__device__ __forceinline__ int edst(const int* ei, int e) { return e < NE ? ei[NE + e] : e - NE; }

// =====================================================================
// fp32 GEMM via V_WMMA_F32_16X16X4_F32.  C[M,NT] = A[M,KT] @ B[KT,NT].
// One wave computes an MT x 1 strip of 16x16 tiles sharing one B
// fragment per K-step.  KT/NT are compile-time so the whole K loop
// unrolls; a 2-deep software pipeline loads step i+1's fragments into a
// distinct register set while step i's MT WMMAs execute (breaks the
// load->wait->wmma serialization seen with a single shared temp).
// OOB N columns are clamped (computed but never stored): no EXEC
// predication anywhere in the hot loop.
// =====================================================================
template <int MT, int KT, int NT>
__global__ __launch_bounds__(256) void gemm_wmma_f32(
    const float* __restrict__ A, const float* __restrict__ B,
    float* __restrict__ C, int M) {
  constexpr int NTILES = (NT + 15) / 16;
  constexpr int NSTEP = KT / 4;
  const int w = blockIdx.x * (blockDim.x >> 5) + (threadIdx.x >> 5);
  const int lane = threadIdx.x & 31;
  const int nStrips = M / (16 * MT);
  if (w >= nStrips * NTILES) return;   // wave-uniform exit: EXEC all-1s below
  const int strip = w / NTILES, tn = w % NTILES;
  const int lmod = lane & 15;
  const int khalf = (lane >> 4) * 2;   // upper half-wave holds K+2,K+3
  const int col = tn * 16 + lmod;
  const bool colOK = (col < NT);
  const int colc = colOK ? col : (NT - 1);

  const float* Abase = A + (size_t)(strip * MT * 16 + lmod) * KT + khalf;
  const float* Bbase = B + (size_t)khalf * NT + colc;

  v8f acc[MT];
#pragma unroll
  for (int t = 0; t < MT; ++t) acc[t] = (v8f){};

  v2f aF[2][MT];
  v2f bF[2];

  // prologue: fragments for K-step 0
  bF[0][0] = Bbase[0];
  bF[0][1] = Bbase[NT];
#pragma unroll
  for (int t = 0; t < MT; ++t) {
    const float* ap = Abase + (size_t)(t * 16) * KT;
    aF[0][t][0] = ap[0];
    aF[0][t][1] = ap[1];
  }

#pragma unroll
  for (int i = 0; i < NSTEP; ++i) {
    const int cur = i & 1, nxt = cur ^ 1;
    if (i + 1 < NSTEP) {               // prefetch next step's fragments
      const int k = (i + 1) * 4;
      bF[nxt][0] = Bbase[(size_t)k * NT];
      bF[nxt][1] = Bbase[(size_t)k * NT + NT];
#pragma unroll
      for (int t = 0; t < MT; ++t) {
        const float* ap = Abase + (size_t)(t * 16) * KT + k;
        aF[nxt][t][0] = ap[0];
        aF[nxt][t][1] = ap[1];
      }
    }
#pragma unroll
    for (int t = 0; t < MT; ++t)
      acc[t] = __builtin_amdgcn_wmma_f32_16x16x4_f32(
          false, aF[cur][t], false, bF[cur], (short)0, acc[t], false, false);
  }

  if (colOK) {
    const int rhalf = (lane >> 4) << 3;  // lanes 16-31 hold M = r+8
#pragma unroll
    for (int t = 0; t < MT; ++t) {
      float* cp = C + (size_t)((strip * MT + t) * 16 + rhalf) * NT + col;
#pragma unroll
      for (int r = 0; r < 8; ++r) cp[(size_t)r * NT] = acc[t][r];
    }
  }
}

// ---------------------------------------------------------------------
__global__ void zero_u32(unsigned* __restrict__ p, int n) {
  int i = blockIdx.x * blockDim.x + threadIdx.x;
  if (i < n) p[i] = 0u;
}

// per-node attention logits: alpha[n,h] = sum_d h[n,h*D+d] * aw[h*D+d]
__global__ void compute_alpha(const float* __restrict__ hproj,
                              const float* __restrict__ aw_src,
                              const float* __restrict__ aw_dst,
                              float* __restrict__ asrc, float* __restrict__ adst,
                              int H, int D) {
  int id = blockIdx.x * blockDim.x + threadIdx.x;
  if (id >= NN * H) return;
  int n = id / H, h = id % H;
  const float* hp = hproj + (size_t)n * (H * D) + h * D;
  const float* ws = aw_src + h * D;
  const float* wd = aw_dst + h * D;
  float s = 0.0f, d = 0.0f;
  for (int j = 0; j < D; ++j) { s += hp[j] * ws[j]; d += hp[j] * wd[j]; }
  asrc[id] = s; adst[id] = d;
}

// pass 1: segment max of leaky-relu'd edge logits
__global__ void edge_max(const int* __restrict__ ei,
                         const float* __restrict__ asrc,
                         const float* __restrict__ adst,
                         unsigned* __restrict__ emax, int H) {
  int e = blockIdx.x * blockDim.x + threadIdx.x;
  if (e >= ET) return;
  int s = esrc(ei, e), d = edst(ei, e);
  for (int h = 0; h < H; ++h) {
    float v = asrc[s * H + h] + adst[d * H + h];
    v = v > 0.0f ? v : NEG_SLOPE * v;
    atomicMax(&emax[d * H + h], f2ord(v));
  }
}

// pass 2: exp(e - max[dst]) ; store per-edge and segment-sum denom
__global__ void edge_expsum(const int* __restrict__ ei,
                            const float* __restrict__ asrc,
                            const float* __restrict__ adst,
                            const unsigned* __restrict__ emax,
                            float* __restrict__ eexp,
                            float* __restrict__ denom, int H) {
  int e = blockIdx.x * blockDim.x + threadIdx.x;
  if (e >= ET) return;
  int s = esrc(ei, e), d = edst(ei, e);
  for (int h = 0; h < H; ++h) {
    float v = asrc[s * H + h] + adst[d * H + h];
    v = v > 0.0f ? v : NEG_SLOPE * v;
    float ex = __expf(v - ord2f(emax[d * H + h]));
    eexp[e * H + h] = ex;
    atomicAdd(&denom[d * H + h], ex);
  }
}

// pass 3: out[dst] += h[src] * alpha   (thread = (edge, head, dim-chunk))
__global__ void edge_aggregate(const int* __restrict__ ei,
                               const float* __restrict__ hproj,
                               const float* __restrict__ eexp,
                               const float* __restrict__ denom,
                               float* __restrict__ outp,
                               int H, int D, int nChunk, int chunkD) {
  int id = blockIdx.x * blockDim.x + threadIdx.x;
  int total = ET * H * nChunk;
  if (id >= total) return;
  int chunk = id % nChunk;
  int t = id / nChunk;
  int h = t % H;
  int e = t / H;
  int s = esrc(ei, e), d = edst(ei, e);
  float alpha = eexp[e * H + h] / (denom[d * H + h] + 1e-16f);
  int base = h * D + chunk * chunkD;
  const float* hp = hproj + (size_t)s * (H * D) + base;
  float* op = outp + (size_t)d * (H * D) + base;
  for (int j = 0; j < chunkD; ++j) atomicAdd(&op[j], hp[j] * alpha);
}

__global__ void bias_relu(float* __restrict__ o, const float* __restrict__ b) {
  int i = blockIdx.x * blockDim.x + threadIdx.x;
  if (i >= NN * HID) return;
  float v = o[i] + b[i & (HID - 1)];
  o[i] = fmaxf(v, 0.0f);
}

__global__ void bias_logsoftmax(const float* __restrict__ o,
                                const float* __restrict__ b,
                                float* __restrict__ out) {
  int n = blockIdx.x * blockDim.x + threadIdx.x;
  if (n >= NN) return;
  float vals[OUTD];
  float mx = -3.402823466e38f;
  for (int c = 0; c < OUTD; ++c) {
    vals[c] = o[(size_t)n * OUTD + c] + b[c];
    mx = fmaxf(mx, vals[c]);
  }
  float s = 0.0f;
  for (int c = 0; c < OUTD; ++c) s += __expf(vals[c] - mx);
  float lse = mx + __logf(s);
  for (int c = 0; c < OUTD; ++c) out[(size_t)n * OUTD + c] = vals[c] - lse;
}

// =====================================================================
extern "C" void kernel_launch(void* const* d_in, const int* in_sizes, int n_in,
                              void* d_out, int out_size, void* d_ws, size_t ws_size,
                              hipStream_t stream) {
  const float* x   = (const float*)d_in[0];
  const int*   ei  = (const int*)d_in[1];
  const float* W1  = (const float*)d_in[2];
  const float* aS1 = (const float*)d_in[3];
  const float* aD1 = (const float*)d_in[4];
  const float* b1  = (const float*)d_in[5];
  const float* W2  = (const float*)d_in[6];
  const float* aS2 = (const float*)d_in[7];
  const float* aD2 = (const float*)d_in[8];
  const float* b2  = (const float*)d_in[9];
  float* dout = (float*)d_out;

  char* ws = (char*)d_ws;
  size_t off = 0;
  auto carve = [&](size_t bytes) {
    char* p = ws + off;
    off += (bytes + 255) & ~(size_t)255;
    return p;
  };
  float*    h1     = (float*)carve((size_t)NN * HID * 4);
  float*    out1   = (float*)carve((size_t)NN * HID * 4);
  float*    h2     = (float*)carve((size_t)NN * OUTD * 4);
  float*    out2   = (float*)carve((size_t)NN * OUTD * 4);
  float*    asrc1  = (float*)carve((size_t)NN * 4 * 4);
  float*    adst1  = (float*)carve((size_t)NN * 4 * 4);
  unsigned* emax1  = (unsigned*)carve((size_t)NN * 4 * 4);
  float*    den1   = (float*)carve((size_t)NN * 4 * 4);
  float*    asrc2  = (float*)carve((size_t)NN * 4);
  float*    adst2  = (float*)carve((size_t)NN * 4);
  unsigned* emax2  = (unsigned*)carve((size_t)NN * 4);
  float*    den2   = (float*)carve((size_t)NN * 4);
  float*    eexp1  = (float*)carve((size_t)ET * 4 * 4);
  float*    eexp2  = (float*)carve((size_t)ET * 4);

  const int T = 256;
  auto blk = [&](long n) { return (int)((n + T - 1) / T); };

  // M/16 = 3125 = 625 strips of 5 tiles
  constexpr int MT = 5;

  // ---------------- layer 1 ----------------
  {
    long waves = (NN / (16 * MT)) * (HID / 16);          // 625 * 4
    gemm_wmma_f32<MT, 128, HID><<<blk(waves * 32), T, 0, stream>>>(x, W1, h1, NN);
  }
  compute_alpha<<<blk((long)NN * 4), T, 0, stream>>>(h1, aS1, aD1, asrc1, adst1, 4, 16);
  zero_u32<<<blk((long)NN * 4), T, 0, stream>>>(emax1, NN * 4);
  zero_u32<<<blk((long)NN * 4), T, 0, stream>>>((unsigned*)den1, NN * 4);
  zero_u32<<<blk((long)NN * HID), T, 0, stream>>>((unsigned*)out1, NN * HID);
  edge_max<<<blk(ET), T, 0, stream>>>(ei, asrc1, adst1, emax1, 4);
  edge_expsum<<<blk(ET), T, 0, stream>>>(ei, asrc1, adst1, emax1, eexp1, den1, 4);
  edge_aggregate<<<blk((long)ET * 4), T, 0, stream>>>(ei, h1, eexp1, den1, out1,
                                                      4, 16, /*nChunk=*/1, /*chunkD=*/16);
  bias_relu<<<blk((long)NN * HID), T, 0, stream>>>(out1, b1);

  // ---------------- layer 2 ----------------
  {
    long waves = (NN / (16 * MT)) * ((OUTD + 15) / 16);  // 625 * 3
    gemm_wmma_f32<MT, HID, OUTD><<<blk(waves * 32), T, 0, stream>>>(out1, W2, h2, NN);
  }
  compute_alpha<<<blk(NN), T, 0, stream>>>(h2, aS2, aD2, asrc2, adst2, 1, OUTD);
  zero_u32<<<blk(NN), T, 0, stream>>>(emax2, NN);
  zero_u32<<<blk(NN), T, 0, stream>>>((unsigned*)den2, NN);
  zero_u32<<<blk((long)NN * OUTD), T, 0, stream>>>((unsigned*)out2, NN * OUTD);
  edge_max<<<blk(ET), T, 0, stream>>>(ei, asrc2, adst2, emax2, 1);
  edge_expsum<<<blk(ET), T, 0, stream>>>(ei, asrc2, adst2, emax2, eexp2, den2, 1);
  edge_aggregate<<<blk((long)ET * 5), T, 0, stream>>>(ei, h2, eexp2, den2, out2,
                                                      1, OUTD, /*nChunk=*/5, /*chunkD=*/8);
  bias_logsoftmax<<<blk(NN), T, 0, stream>>>(out2, b2, dout);
}